// LSTM_87823491269162
// MI455X (gfx1250) — compile-verified
//
#include <hip/hip_runtime.h>
#include <hip/hip_bf16.h>

// Problem constants (match reference)
#define T_  512
#define B_  64
#define I_  1024
#define H_  1024
#define G4_ 4096   // 4*H

typedef __attribute__((ext_vector_type(2))) float v2f;
typedef __attribute__((ext_vector_type(8))) float v8f;

__device__ __forceinline__ float sigmoidf_(float x) {
    return 1.0f / (1.0f + __expf(-x));
}

// ---------------------------------------------------------------------------
// B-matrix LDS layout: k-pair interleaved. Pair-row p holds, for every column
// n, the float2 {B[2p][n], B[2p+1][n]} at offset p*LDBP + 2n. A WMMA B
// fragment for lane L (half = L/16) at K-step ks is then a single aligned
// 8-byte ds_load_b64 at pair-row ks/2 + half -> no register shuffles.
//
// Global->LDS staging uses register prefetch double-buffering: thread-private
// float4 registers hold chunk k+1 while WMMAs consume chunk k from LDS, so
// global_load_b128 latency overlaps compute and loads issue in a batch.
// ---------------------------------------------------------------------------

// Phase 1: XG[T*B, 4H] = X[T*B, I] @ W_ih^T + b_ih + b_hh
// Block: 256 threads (8 waves), tile 64(M) x 64(N), K-chunk 32.
#define LDA1  36    // 32 + 4 pad (floats)
#define LDBP1 132   // pair-row stride: 2*64 + 4 (floats)

__global__ __launch_bounds__(256) void lstm_xgates_gemm(
    const float* __restrict__ X,    // [T*B, I]
    const float* __restrict__ Wih,  // [4H, I]
    const float* __restrict__ bih,  // [4H]
    const float* __restrict__ bhh,  // [4H]
    float* __restrict__ XG)         // [T*B, 4H]
{
    __shared__ __align__(16) float Ash[64 * LDA1];    // A: 64 rows x 32 k
    __shared__ __align__(16) float Bsh[16 * LDBP1];   // B: 16 k-pairs x 64 n (float2 cells)

    const int tid  = threadIdx.x;
    const int lane = tid & 31;
    const int wave = tid >> 5;
    const int wm   = wave >> 1;        // 0..3
    const int wn   = wave & 1;         // 0..1
    const int half = lane >> 4;        // 0..1
    const int l16  = lane & 15;        // 0..15
    const int m0   = blockIdx.y * 64;
    const int n0   = blockIdx.x * 64;

    // Per-thread staging assignment: row = tid/8 (and +32), kq = tid%8.
    const int srow = tid >> 3, skq = tid & 7;
    const float* pA0 = X   + (size_t)(m0 + srow)      * I_ + skq * 4;
    const float* pA1 = X   + (size_t)(m0 + srow + 32) * I_ + skq * 4;
    const float* pB0 = Wih + (size_t)(n0 + srow)      * I_ + skq * 4;
    const float* pB1 = Wih + (size_t)(n0 + srow + 32) * I_ + skq * 4;

    // Prefetch chunk 0 into registers.
    float4 ra0 = *(const float4*)(pA0);
    float4 ra1 = *(const float4*)(pA1);
    float4 rb0 = *(const float4*)(pB0);
    float4 rb1 = *(const float4*)(pB1);

    v8f acc0 = {};
    v8f acc1 = {};

    for (int k0 = 0; k0 < I_; k0 += 32) {
        // --- commit staged registers to LDS ---
        *(float4*)&Ash[srow * LDA1 + skq * 4]        = ra0;
        *(float4*)&Ash[(srow + 32) * LDA1 + skq * 4] = ra1;
        {
            v2f q;
            q.x = rb0.x; q.y = rb0.y; *(v2f*)&Bsh[(skq * 2 + 0) * LDBP1 + 2 * srow] = q;
            q.x = rb0.z; q.y = rb0.w; *(v2f*)&Bsh[(skq * 2 + 1) * LDBP1 + 2 * srow] = q;
            q.x = rb1.x; q.y = rb1.y; *(v2f*)&Bsh[(skq * 2 + 0) * LDBP1 + 2 * (srow + 32)] = q;
            q.x = rb1.z; q.y = rb1.w; *(v2f*)&Bsh[(skq * 2 + 1) * LDBP1 + 2 * (srow + 32)] = q;
        }
        __syncthreads();

        // --- issue prefetch of next chunk (overlaps the WMMA sequence) ---
        if (k0 + 32 < I_) {
            ra0 = *(const float4*)(pA0 + k0 + 32);
            ra1 = *(const float4*)(pA1 + k0 + 32);
            rb0 = *(const float4*)(pB0 + k0 + 32);
            rb1 = *(const float4*)(pB1 + k0 + 32);
        }

        // --- 8 WMMA K-steps of 4 ---
        #pragma unroll
        for (int ks = 0; ks < 32; ks += 4) {
            const int ka = ks + 2 * half;            // lane-pair K base (ISA A layout)
            const int kp = (ks >> 1) + half;         // B pair-row for this lane
            const v2f a  = *(const v2f*)&Ash[(wm * 16 + l16) * LDA1 + ka];
            const v2f b0 = *(const v2f*)&Bsh[kp * LDBP1 + 2 * (wn * 32 + l16)];
            const v2f b1 = *(const v2f*)&Bsh[kp * LDBP1 + 2 * (wn * 32 + 16 + l16)];
            acc0 = __builtin_amdgcn_wmma_f32_16x16x4_f32(false, a, false, b0,
                                                         (short)0, acc0, false, false);
            acc1 = __builtin_amdgcn_wmma_f32_16x16x4_f32(false, a, false, b1,
                                                         (short)0, acc1, false, false);
        }
        __syncthreads();
    }

    // --- epilogue: add biases, store non-temporally (536 MB stream, read once
    //     much later by phase 2; keep it out of L2 so the weights stay hot) ---
    const int col = n0 + wn * 32 + l16;
    const float bsum0 = bih[col] + bhh[col];
    const float bsum1 = bih[col + 16] + bhh[col + 16];
    #pragma unroll
    for (int r = 0; r < 8; ++r) {
        const int row = m0 + wm * 16 + r + 8 * half;
        __builtin_nontemporal_store(acc0[r] + bsum0, &XG[(size_t)row * G4_ + col]);
        __builtin_nontemporal_store(acc1[r] + bsum1, &XG[(size_t)row * G4_ + col + 16]);
    }
}

// ---------------------------------------------------------------------------
// Phase 2: one timestep. gates = XG[t] + h_prev @ W_hh^T ; fused LSTM update.
// Block: 256 threads, all 64 batch rows, a 32-wide H slice for ALL four gates.
// Wave (wm 0..3, wn 0..1): one 16x16 tile position, 4 gate accumulators.
// ---------------------------------------------------------------------------
#define LDA2  36
#define LDBP2 68    // pair-row stride: 2*32 + 4 (floats)

__global__ __launch_bounds__(256) void lstm_step(
    const float* __restrict__ hprev,  // [B, H]  (initial_h or out[t-1])
    const float* __restrict__ Whh,    // [4H, H] (L2-resident: 16 MB << 192 MB)
    const float* __restrict__ XGt,    // [B, 4H] slice for this t
    const float* __restrict__ cprev,  // [B, H]  (initial_c or cbuf)
    float* __restrict__ cbuf,         // [B, H]
    float* __restrict__ hout,         // [B, H]  = out[t]
    float* __restrict__ hfin,         // final-h slot or nullptr
    float* __restrict__ cfin)         // final-c slot or nullptr
{
    __shared__ __align__(16) float Ash[64 * LDA2];        // h tile: 64 x 32
    __shared__ __align__(16) float Bsh[4][16 * LDBP2];    // per gate: 16 k-pairs x 32 n

    const int tid  = threadIdx.x;
    const int lane = tid & 31;
    const int wave = tid >> 5;
    const int wm   = wave >> 1;       // 0..3  (batch rows wm*16..)
    const int wn   = wave & 1;        // 0..1  (n offset wn*16)
    const int half = lane >> 4;
    const int l16  = lane & 15;
    const int n0   = blockIdx.x * 32; // H-slice base

    // Per-thread staging: A rows tid/8 (+32); B row tid/8 for each of 4 gates.
    const int srow = tid >> 3, skq = tid & 7;
    const float* pA0 = hprev + (size_t)srow        * H_ + skq * 4;
    const float* pA1 = hprev + (size_t)(srow + 32) * H_ + skq * 4;
    const float* pBg[4];
    #pragma unroll
    for (int g = 0; g < 4; ++g)
        pBg[g] = Whh + (size_t)(g * H_ + n0 + srow) * H_ + skq * 4;

    float4 ra0 = *(const float4*)(pA0);
    float4 ra1 = *(const float4*)(pA1);
    float4 rb[4];
    #pragma unroll
    for (int g = 0; g < 4; ++g) rb[g] = *(const float4*)(pBg[g]);

    v8f acc[4] = {{}, {}, {}, {}};

    for (int k0 = 0; k0 < H_; k0 += 32) {
        // --- commit staged registers to LDS ---
        *(float4*)&Ash[srow * LDA2 + skq * 4]        = ra0;
        *(float4*)&Ash[(srow + 32) * LDA2 + skq * 4] = ra1;
        #pragma unroll
        for (int g = 0; g < 4; ++g) {
            v2f q;
            q.x = rb[g].x; q.y = rb[g].y; *(v2f*)&Bsh[g][(skq * 2 + 0) * LDBP2 + 2 * srow] = q;
            q.x = rb[g].z; q.y = rb[g].w; *(v2f*)&Bsh[g][(skq * 2 + 1) * LDBP2 + 2 * srow] = q;
        }
        __syncthreads();

        // --- prefetch next chunk ---
        if (k0 + 32 < H_) {
            ra0 = *(const float4*)(pA0 + k0 + 32);
            ra1 = *(const float4*)(pA1 + k0 + 32);
            #pragma unroll
            for (int g = 0; g < 4; ++g) rb[g] = *(const float4*)(pBg[g] + k0 + 32);
        }

        #pragma unroll
        for (int ks = 0; ks < 32; ks += 4) {
            const int ka = ks + 2 * half;
            const int kp = (ks >> 1) + half;
            const v2f a = *(const v2f*)&Ash[(wm * 16 + l16) * LDA2 + ka];
            #pragma unroll
            for (int g = 0; g < 4; ++g) {
                const v2f b = *(const v2f*)&Bsh[g][kp * LDBP2 + 2 * (wn * 16 + l16)];
                acc[g] = __builtin_amdgcn_wmma_f32_16x16x4_f32(false, a, false, b,
                                                               (short)0, acc[g], false, false);
            }
        }
        __syncthreads();
    }

    // --- fused LSTM epilogue (XG stream read once -> non-temporal loads) ---
    const int h = n0 + wn * 16 + l16;   // H column
    #pragma unroll
    for (int r = 0; r < 8; ++r) {
        const int b = wm * 16 + r + 8 * half;        // batch row (M=64 fully covered)
        const float* xg = XGt + (size_t)b * G4_;
        const float iv = sigmoidf_(acc[0][r] + __builtin_nontemporal_load(&xg[h]));
        const float fv = sigmoidf_(acc[1][r] + __builtin_nontemporal_load(&xg[H_ + h]));
        const float gv = tanhf    (acc[2][r] + __builtin_nontemporal_load(&xg[2 * H_ + h]));
        const float ov = sigmoidf_(acc[3][r] + __builtin_nontemporal_load(&xg[3 * H_ + h]));
        const float cn = fv * cprev[(size_t)b * H_ + h] + iv * gv;
        const float hn = ov * tanhf(cn);
        cbuf[(size_t)b * H_ + h] = cn;
        hout[(size_t)b * H_ + h] = hn;
        if (hfin) {                       // uniform branch, last timestep only
            hfin[(size_t)b * H_ + h] = hn;
            cfin[(size_t)b * H_ + h] = cn;
        }
    }
}

// ---------------------------------------------------------------------------
extern "C" void kernel_launch(void* const* d_in, const int* in_sizes, int n_in,
                              void* d_out, int out_size, void* d_ws, size_t ws_size,
                              hipStream_t stream) {
    const float* X   = (const float*)d_in[0];  // [T, B, I]
    const float* h0  = (const float*)d_in[1];  // [B, H]
    const float* c0  = (const float*)d_in[2];  // [B, H]
    const float* Wih = (const float*)d_in[3];  // [4H, I]
    const float* Whh = (const float*)d_in[4];  // [4H, H]
    const float* bih = (const float*)d_in[5];  // [4H]
    const float* bhh = (const float*)d_in[6];  // [4H]
    float* out = (float*)d_out;                // [T*B*H] outputs | [B*H] h | [B*H] c

    float* XG   = (float*)d_ws;                         // [T*B, 4H]
    float* cbuf = XG + (size_t)T_ * B_ * G4_;           // [B, H]

    // Phase 1: big memory-bound GEMM for all input-side gate projections.
    dim3 g1(G4_ / 64, (T_ * B_) / 64);                  // (64, 512)
    lstm_xgates_gemm<<<g1, 256, 0, stream>>>(X, Wih, bih, bhh, XG);

    // Phase 2: 512 serial steps; h state flows through the output tensor.
    for (int t = 0; t < T_; ++t) {
        const float* hp = (t == 0) ? h0 : out + (size_t)(t - 1) * B_ * H_;
        const float* cp = (t == 0) ? c0 : cbuf;
        float* hfin = (t == T_ - 1) ? out + (size_t)T_ * B_ * H_ : nullptr;
        float* cfin = (t == T_ - 1) ? out + (size_t)T_ * B_ * H_ + (size_t)B_ * H_ : nullptr;
        lstm_step<<<dim3(H_ / 32), 256, 0, stream>>>(
            hp, Whh, XG + (size_t)t * B_ * G4_, cp, cbuf,
            out + (size_t)t * B_ * H_, hfin, cfin);
    }
}